// AveragedHausdorffLoss_3753801417055
// MI455X (gfx1250) — compile-verified
//
#include <hip/hip_runtime.h>
#include <hip/hip_bf16.h>
#include <math.h>

typedef __attribute__((ext_vector_type(2))) float v2f;
typedef __attribute__((ext_vector_type(8))) float v8f;

#define WAVE 32
#define INF_BITS 0x7F800000u
#define MAX_M_LDS 8192   // column-min staging in LDS (32 KB); M==8192 here

// ---------------------------------------------------------------------------
// Kernel 0: initialize workspace (col-min = +inf, row accumulators = 0)
// ---------------------------------------------------------------------------
__global__ void hd_init(unsigned* __restrict__ colmin, float* __restrict__ rowsum,
                        unsigned* __restrict__ rowmax, int M) {
    int i = blockIdx.x * blockDim.x + threadIdx.x;
    if (i < M) colmin[i] = INF_BITS;
    if (i == 0) { rowsum[0] = 0.0f; rowmax[0] = 0u; }
}

// ---------------------------------------------------------------------------
// Kernel 1: pairwise d^2 via V_WMMA_F32_16X16X4_F32, fused min-reductions.
//   d^2 = ||a||^2 - 2 a.b + ||b||^2  ==  A(16x4) x B(4x16) + C(16x16)
//   A row = [-2x, -2y, -2z, 1],  B col = [x, y, z, ||b||^2],  C = ||a||^2 bcast
// Each wave owns one 16-row tile of set1 and iterates all column tiles.
// B fetch is software-pipelined (prefetch next tile before consuming current)
// so the s_wait_loadcnt overlaps the WMMA + min-chain of the current tile.
// ---------------------------------------------------------------------------
__global__ __launch_bounds__(256) void hd_main(const float* __restrict__ s1,
                                               const float* __restrict__ s2,
                                               int N, int M,
                                               unsigned* __restrict__ g_colmin,
                                               float* __restrict__ g_rowsum,
                                               unsigned* __restrict__ g_rowmax) {
    __shared__ unsigned s_colmin[MAX_M_LDS];

    const int tid = threadIdx.x;
    for (int i = tid; i < M; i += blockDim.x) s_colmin[i] = INF_BITS;
    __syncthreads();

    const int lane = tid & (WAVE - 1);
    const int wave = tid >> 5;                 // 8 waves per block
    const int rowTile = blockIdx.x * 8 + wave; // 16 rows of set1 per wave
    const int r0 = rowTile * 16;
    const bool validTile = (r0 < N);
    const int hi = (lane >> 4) & 1;            // lane half selects K pair / M+8
    const int m  = lane & 15;

    // ---- A fragment: 16x4 fp32, 2 VGPRs.
    // lanes 0-15: (K=0,K=1) = (-2x, -2y); lanes 16-31: (K=2,K=3) = (-2z, 1)
    int arow = min(r0 + m, N - 1);
    float ax = s1[arow * 3 + 0];
    float ay = s1[arow * 3 + 1];
    float az = s1[arow * 3 + 2];
    float n2a = ax * ax + ay * ay + az * az;   // ||a||^2 for row r0+m
    v2f afrag;
    afrag[0] = hi ? (-2.0f * az) : (-2.0f * ax);
    afrag[1] = hi ? 1.0f         : (-2.0f * ay);

    // ---- C fragment: ||a_M||^2 broadcast along row M.
    // C layout: VGPR v, lanes 0-15 -> M=v, lanes 16-31 -> M=v+8.
    v8f cfrag;
#pragma unroll
    for (int v = 0; v < 8; ++v)
        cfrag[v] = __shfl(n2a, v + hi * 8, WAVE);   // row (v + hi*8)'s norm

    // Running row-min of d^2, kept in the D-matrix register layout.
    v8f rmin;
#pragma unroll
    for (int v = 0; v < 8; ++v) rmin[v] = __builtin_inff();

    const int nColTiles = (M + 15) >> 4;

    // ---- Software-pipelined B fetch: prologue loads tile 0.
    int bcol = min(m, M - 1);
    float bx = s2[bcol * 3 + 0];
    float by = s2[bcol * 3 + 1];
    float bz = s2[bcol * 3 + 2];

    for (int j = 0; j < nColTiles; ++j) {
        // Issue next tile's loads NOW; consumed at loop bottom, so the wait
        // overlaps this iteration's WMMA + reduction work.
        int nxtcol = min((j + 1) * 16 + m, M - 1);
        float nx = s2[nxtcol * 3 + 0];
        float ny = s2[nxtcol * 3 + 1];
        float nz = s2[nxtcol * 3 + 2];

        // ---- B fragment: 4x16 fp32, 2 VGPRs.
        // lanes 0-15: (K=0,K=1) = (x, y); lanes 16-31: (K=2,K=3) = (z, ||b||^2)
        float n2b = bx * bx + by * by + bz * bz;
        v2f bfrag;
        bfrag[0] = hi ? bz  : bx;
        bfrag[1] = hi ? n2b : by;

        // D = A x B + C  (one 16x16 tile of squared distances)
        v8f d = __builtin_amdgcn_wmma_f32_16x16x4_f32(
            /*neg_a=*/false, afrag, /*neg_b=*/false, bfrag,
            /*c_mod=*/(short)0, cfrag, /*reuse_a=*/false, /*reuse_b=*/false);

        // Row-min accumulate (stays in D register layout until epilogue).
#pragma unroll
        for (int v = 0; v < 8; ++v) rmin[v] = fminf(rmin[v], d[v]);

        // Column partial min over this lane's 8 rows. Both lane-halves write
        // their partial to the same LDS word (lo half: rows 0-7, hi half:
        // rows 8-15); the LDS atomic unit merges them. No EXEC manipulation,
        // no cross-lane shuffle needed.
        float cp = fminf(fminf(fminf(d[0], d[1]), fminf(d[2], d[3])),
                         fminf(fminf(d[4], d[5]), fminf(d[6], d[7])));
        cp = fmaxf(cp, 0.0f);                 // clamp FP cancellation < 0
        atomicMin(&s_colmin[j * 16 + m], __float_as_uint(cp));

        bx = nx; by = ny; bz = nz;            // rotate pipeline
    }

    // ---- Row-min: reduce across the 16-lane half (masks stay inside a half).
#pragma unroll
    for (int mask = 1; mask <= 8; mask <<= 1) {
#pragma unroll
        for (int v = 0; v < 8; ++v)
            rmin[v] = fminf(rmin[v], __shfl_xor(rmin[v], mask, WAVE));
    }
    if (validTile && m == 0) {   // lane 0 -> rows r0..r0+7, lane 16 -> +8..+15
        float sum = 0.0f, mx = 0.0f;
#pragma unroll
        for (int v = 0; v < 8; ++v) {
            int row = r0 + v + hi * 8;
            if (row < N) {
                float d2 = fmaxf(rmin[v], 0.0f);
                sum += sqrtf(d2);
                mx = fmaxf(mx, d2);
            }
        }
        atomicAdd(g_rowsum, sum);                       // global_atomic_add_f32
        atomicMax(g_rowmax, __float_as_uint(mx));       // d^2; sqrt at the end
    }

    __syncthreads();
    for (int i = tid; i < M; i += blockDim.x)
        atomicMin(&g_colmin[i], s_colmin[i]);           // global_atomic_min_u32
}

// ---------------------------------------------------------------------------
// Kernel 2: reduce column mins, combine terms, honor flags from device memory
// ---------------------------------------------------------------------------
__global__ void hd_final(const unsigned* __restrict__ colmin,
                         const float* __restrict__ rowsum,
                         const unsigned* __restrict__ rowmax,
                         const int* __restrict__ hflag,
                         const int* __restrict__ w12p,
                         const int* __restrict__ w21p,
                         float* __restrict__ out,
                         int N, int M, int out_size) {
    __shared__ float s_sum[256];
    __shared__ float s_max[256];
    const int tid = threadIdx.x;
    float sum = 0.0f, mx = 0.0f;
    for (int i = tid; i < M; i += blockDim.x) {
        float d2 = fmaxf(__uint_as_float(colmin[i]), 0.0f);
        sum += sqrtf(d2);
        mx = fmaxf(mx, d2);
    }
    s_sum[tid] = sum;
    s_max[tid] = mx;
    __syncthreads();
    for (int s = 128; s > 0; s >>= 1) {
        if (tid < s) {
            s_sum[tid] += s_sum[tid + s];
            s_max[tid] = fmaxf(s_max[tid], s_max[tid + s]);
        }
        __syncthreads();
    }
    if (tid == 0) {
        const int h = hflag[0];
        float t12 = 0.0f, t21 = 0.0f;
        if (w12p[0] != 0)
            t12 = h ? sqrtf(fmaxf(__uint_as_float(rowmax[0]), 0.0f))
                    : rowsum[0] / (float)N;
        if (w21p[0] != 0)
            t21 = h ? sqrtf(s_max[0]) : s_sum[0] / (float)M;
        if (out_size >= 2) { out[0] = t12; out[1] = t21; }
        else               { out[0] = t12 + t21; }
    }
}

// ---------------------------------------------------------------------------
extern "C" void kernel_launch(void* const* d_in, const int* in_sizes, int n_in,
                              void* d_out, int out_size, void* d_ws, size_t ws_size,
                              hipStream_t stream) {
    const float* s1  = (const float*)d_in[0];
    const float* s2  = (const float*)d_in[1];
    const int* hflag = (const int*)d_in[2];
    const int* w12   = (const int*)d_in[3];
    const int* w21   = (const int*)d_in[4];

    const int N = in_sizes[0] / 3;
    const int M = in_sizes[1] / 3;

    // Workspace layout: [M x u32 colmin][f32 rowsum][u32 rowmax]
    unsigned* colmin = (unsigned*)d_ws;
    float*    rowsum = (float*)((char*)d_ws + (size_t)M * sizeof(unsigned));
    unsigned* rowmax = (unsigned*)((char*)d_ws + (size_t)M * sizeof(unsigned) + sizeof(float));

    hd_init<<<(M + 255) / 256, 256, 0, stream>>>(colmin, rowsum, rowmax, M);

    const int rowTiles = (N + 15) / 16;        // 512 for N=8192
    const int blocks   = (rowTiles + 7) / 8;   // 8 waves (row tiles) per block
    hd_main<<<blocks, 256, 0, stream>>>(s1, s2, N, M, colmin, rowsum, rowmax);

    hd_final<<<1, 256, 0, stream>>>(colmin, rowsum, rowmax, hflag, w12, w21,
                                    (float*)d_out, N, M, out_size);
}